// LXUCacheEmbeddingBag_83915071029440
// MI455X (gfx1250) — compile-verified
//
#include <hip/hip_runtime.h>

typedef __attribute__((ext_vector_type(2))) float v2f;
typedef __attribute__((ext_vector_type(8))) float v8f;

#define EMBED_DIM 128
#define TILES 8          // 128 dims / 16 per WMMA tile
#define BAGS_PER_BLOCK 8 // 8 waves of 32

__global__ __launch_bounds__(256) void embedbag_wmma_kernel(
    const float* __restrict__ weights,
    const int* __restrict__ indices,
    const int* __restrict__ offsets,
    float* __restrict__ out,
    int num_bags)
{
  const int lane = threadIdx.x & 31;
  const int wave = threadIdx.x >> 5;
  const int bag  = blockIdx.x * BAGS_PER_BLOCK + wave;
  if (bag >= num_bags) return;   // wave-uniform: surviving waves keep EXEC all-ones

  const int start = offsets[bag];
  const int end   = offsets[bag + 1];
  const int count = end - start;

  const int ln   = lane & 15;   // N coordinate within the 16-wide tile
  const int half = lane >> 4;   // this half-wave feeds two K slots of the 4-row chunk

  // One 16x16 f32 accumulator per 16-dim tile; every row of D holds the pooled sum.
  v8f acc[TILES];
  const v8f zero8 = {0.f, 0.f, 0.f, 0.f, 0.f, 0.f, 0.f, 0.f};
#pragma unroll
  for (int t = 0; t < TILES; ++t) acc[t] = zero8;

  const v2f a_ones = {1.0f, 1.0f};  // A = 16x4 ones matrix (the reduction operator)

  // ---------------- steady state: full chunks of 4 pooled rows, no masking ----
  const int full = count >> 2;
#pragma unroll 2
  for (int c = 0; c < full; ++c) {
    // lanes 0-15 consume rows j0,j0+1; lanes 16-31 rows j0+2,j0+3 (K-order
    // irrelevant: A is all-ones, so D = column sums of B for any K permutation)
    const int base = start + (c << 2) + (half << 1);
    const int iA = indices[base];
    const int iB = indices[base + 1];
    const float* rA = weights + (long)iA * EMBED_DIM + ln;
    const float* rB = weights + (long)iB * EMBED_DIM + ln;
#pragma unroll
    for (int t = 0; t < TILES; ++t) {
      v2f b;
      b.x = rA[t * 16];            // 64B contiguous per half-wave
      b.y = rB[t * 16];
      acc[t] = __builtin_amdgcn_wmma_f32_16x16x4_f32(
          false, a_ones, false, b, (short)0, acc[t], false, false);
    }
  }

  // ---------------- tail: at most one masked chunk (wave-uniform branch) ------
  const int rem = count & 3;
  if (rem) {
    const int j0 = full << 2;
    const int jA = j0 + (half << 1);
    const int jB = jA + 1;
    const bool vA = jA < count;
    const bool vB = jB < count;
    const int  iA = indices[start + (vA ? jA : 0)];
    const int  iB = indices[start + (vB ? jB : 0)];
    const float* rA = weights + (long)iA * EMBED_DIM + ln;
    const float* rB = weights + (long)iB * EMBED_DIM + ln;
#pragma unroll
    for (int t = 0; t < TILES; ++t) {
      v2f b;
      b.x = vA ? rA[t * 16] : 0.0f;
      b.y = vB ? rB[t * 16] : 0.0f;
      acc[t] = __builtin_amdgcn_wmma_f32_16x16x4_f32(
          false, a_ones, false, b, (short)0, acc[t], false, false);
    }
  }

  // All D rows are identical; lane l holds tile t's column (l&15) in acc[t][0].
  // Low half stores tiles 0-3, high half stores tiles 4-7.
  float* outb = out + (long)bag * EMBED_DIM;
#pragma unroll
  for (int t = 0; t < 4; ++t) {
    const float lo  = acc[t][0];
    const float hi  = acc[t + 4][0];
    const float val = half ? hi : lo;
    const int   d   = t * 16 + ln + half * 64;
    outb[d] = val;
  }
}

extern "C" void kernel_launch(void* const* d_in, const int* in_sizes, int n_in,
                              void* d_out, int out_size, void* d_ws, size_t ws_size,
                              hipStream_t stream) {
  const float* weights = (const float*)d_in[0];
  const int*   indices = (const int*)d_in[1];
  const int*   offsets = (const int*)d_in[2];
  float*       out     = (float*)d_out;

  const int num_bags = in_sizes[2] - 1;   // offsets has num_bags+1 entries
  const int blocks   = (num_bags + BAGS_PER_BLOCK - 1) / BAGS_PER_BLOCK;

  hipLaunchKernelGGL(embedbag_wmma_kernel, dim3(blocks), dim3(256), 0, stream,
                     weights, indices, offsets, out, num_bags);
}